// OrientedTextPostProcessing_67585605370183
// MI455X (gfx1250) — compile-verified
//
#include <hip/hip_runtime.h>
#include <math.h>

// ---------------- feature probes (gfx1250 toolchain) ----------------
#if defined(__has_builtin)
#  if __has_builtin(__builtin_amdgcn_global_load_async_to_lds_b64) && \
      __has_builtin(__builtin_amdgcn_s_wait_asynccnt)
#    define USE_ASYNC_LDS 1
#  endif
#  if __has_builtin(__builtin_amdgcn_s_wait_tensorcnt)
#    define HAVE_TENSORCNT 1
#  endif
#  if __has_builtin(__builtin_amdgcn_wmma_f32_16x16x4_f32)
#    define HAVE_WMMA_F32X4 1
#  endif
#endif
#ifndef USE_ASYNC_LDS
#  define USE_ASYNC_LDS 0
#endif
#ifndef HAVE_TENSORCNT
#  define HAVE_TENSORCNT 0
#endif
#ifndef HAVE_WMMA_F32X4
#  define HAVE_WMMA_F32X4 0
#endif

typedef float v2f __attribute__((ext_vector_type(2)));
typedef float v8f __attribute__((ext_vector_type(8)));
// Matches the builtin's parameter type from the hipcc diagnostic:
// "__attribute__((__vector_size__(2 * sizeof(int)))) int __device__ *"
typedef int v2i_b64 __attribute__((vector_size(2 * sizeof(int))));

#define NPIX   (512 * 512)
#define KW     512
#define KC     2048
#define NCLS   68
#define TILE   2048
#define CAP    4096   // LDS candidate buffer for adaptive radix select

// d_out layout (floats), reference return order:
// char_bboxes(2048x9), char_scores(2048x68), word_bboxes(512x9), wkeep(512), ckeep(2048)
#define OUT_CHAR_BB 0
#define OUT_CHAR_SC (OUT_CHAR_BB + KC * 9)        // 18432
#define OUT_WORD_BB (OUT_CHAR_SC + KC * NCLS)     // 157696
#define OUT_WKEEP   (OUT_WORD_BB + KW * 9)        // 162304
#define OUT_CKEEP   (OUT_WKEEP + KW)              // 162816

// ---------------- kernels ----------------

// Zero suppression matrices + compaction counters; emit a tensorcnt wait as a
// guaranteed gfx1250 marker (counter is 0 -> free).
__global__ void init_kernel(unsigned* __restrict__ supp, int n_supp,
                            unsigned* __restrict__ cnt) {
  int i = blockIdx.x * blockDim.x + threadIdx.x;
  if (i < n_supp) supp[i] = 0u;
  if (i < 2) cnt[i] = 0u;
#if HAVE_TENSORCNT
  __builtin_amdgcn_s_wait_tensorcnt(0);
#endif
}

// Pack (score, ~index) into unique 64-bit keys. Scores are in [0,1) so their
// float bits are monotone as u32; ties resolve to the lowest index (JAX top_k).
__global__ void build_keys_kernel(const float* __restrict__ wfg,
                                  const float* __restrict__ cfg,
                                  unsigned long long* __restrict__ wkeys,
                                  unsigned long long* __restrict__ ckeys) {
  int i = blockIdx.x * blockDim.x + threadIdx.x;
  if (i >= NPIX) return;
  float wf = wfg[i];
  bool wm = wf > 0.5f;
  float wsv = wm ? wf : 0.0f;
  unsigned inv = ~(unsigned)i;
  wkeys[i] = ((unsigned long long)__float_as_uint(wsv) << 32) | inv;
  float cf = cfg[i];
  float csv = (wm && cf > 0.25f) ? cf : 0.0f;
  ckeys[i] = ((unsigned long long)__float_as_uint(csv) << 32) | inv;
}

// Exact 8-pass MSB radix select of the K-th largest 64-bit key (keys unique).
// Full-array passes stage tiles through LDS with CDNA5 async global->LDS
// loads. As soon as the surviving bucket fits in LDS (<= CAP keys), the
// candidates are compacted into LDS and all remaining passes run from LDS,
// cutting global traffic from 8 sweeps to ~3.
__global__ __launch_bounds__(1024) void radix_select_kernel(
    const unsigned long long* __restrict__ keys, int N, int K,
    unsigned long long* __restrict__ out_thresh) {
  __shared__ unsigned long long stage[TILE];
  __shared__ unsigned long long cand[CAP];
  __shared__ unsigned hist[256];
  __shared__ unsigned long long sh_prefix;
  __shared__ int sh_krem;
  __shared__ int sh_cand;    // #candidates in cand[] (-1 => global mode)
  __shared__ int sh_cnt;
  __shared__ int sh_bincnt;  // size of the bucket chosen this pass
  const int t = threadIdx.x;
  if (t == 0) { sh_prefix = 0ull; sh_krem = K; sh_cand = -1; sh_cnt = 0; }
  __syncthreads();
  for (int p = 7; p >= 0; --p) {
    if (t < 256) hist[t] = 0u;
    __syncthreads();
    const unsigned long long prefix = sh_prefix;
    const unsigned long long pmask =
        (p == 7) ? 0ull : (~0ull << (unsigned)((p + 1) * 8));
    const int shift = p * 8;
    const int ncand = sh_cand;
    if (ncand >= 0) {
      // LDS-resident candidate mode
      for (int e = t; e < ncand; e += 1024) {
        unsigned long long k = cand[e];
        if ((k & pmask) == prefix)
          atomicAdd(&hist[(unsigned)(k >> shift) & 0xFFu], 1u);
      }
      __syncthreads();
    } else {
      // Global mode: stream the whole key array through LDS tiles
      for (int base = 0; base < N; base += TILE) {
#if USE_ASYNC_LDS
        for (int e = t; e < TILE; e += 1024) {
          __builtin_amdgcn_global_load_async_to_lds_b64(
              (__attribute__((address_space(1))) v2i_b64*)(keys + base + e),
              (__attribute__((address_space(3))) v2i_b64*)(&stage[e]), 0, 0);
        }
        __builtin_amdgcn_s_wait_asynccnt(0);
        __syncthreads();
#else
        for (int e = t; e < TILE; e += 1024) stage[e] = keys[base + e];
        __syncthreads();
#endif
        for (int e = t; e < TILE; e += 1024) {
          unsigned long long k = stage[e];
          if ((k & pmask) == prefix) {
            unsigned d = (unsigned)(k >> shift) & 0xFFu;
            atomicAdd(&hist[d], 1u);
          }
        }
        __syncthreads();
      }
    }
    if (t == 0) {
      int kr = sh_krem;
      int cum = 0;
      unsigned long long pf = sh_prefix;
      int chosen = 0;
      for (int d = 255; d >= 0; --d) {
        int c = (int)hist[d];
        if (cum + c >= kr) {
          pf |= ((unsigned long long)(unsigned)d) << shift;
          sh_krem = kr - cum;
          chosen = c;
          break;
        }
        cum += c;
      }
      sh_prefix = pf;
      sh_bincnt = chosen;
    }
    __syncthreads();
    // Adaptive compaction: surviving bucket now fits in LDS.
    if (sh_cand < 0 && p > 0 && sh_bincnt <= CAP) {
      const unsigned long long pf = sh_prefix;
      const unsigned long long pm = (~0ull) << (unsigned)shift;
      for (int i = t; i < N; i += 1024) {
        unsigned long long k = keys[i];
        if ((k & pm) == pf) {
          int pos = atomicAdd(&sh_cnt, 1);
          cand[pos] = k;
        }
      }
      __syncthreads();
      if (t == 0) sh_cand = sh_cnt;
      __syncthreads();
    }
  }
  if (t == 0) *out_thresh = sh_prefix;
}

// Gather all keys >= threshold (exactly K because keys are unique).
__global__ void compact_kernel(const unsigned long long* __restrict__ keys,
                               int N,
                               const unsigned long long* __restrict__ thresh,
                               unsigned long long* __restrict__ outsel,
                               unsigned* __restrict__ cnt) {
  int i = blockIdx.x * blockDim.x + threadIdx.x;
  if (i >= N) return;
  unsigned long long T = *thresh;
  unsigned long long k = keys[i];
  if (k >= T) {
    unsigned pos = atomicAdd(cnt, 1u);
    outsel[pos] = k;
  }
}

// In-LDS bitonic sort, descending (K power of two, K <= 2048).
__global__ void bitonic_sort_kernel(unsigned long long* __restrict__ data,
                                    int K) {
  __shared__ unsigned long long s[KC];
  const int t = threadIdx.x, nt = blockDim.x;
  for (int i = t; i < K; i += nt) s[i] = data[i];
  __syncthreads();
  for (int k = 2; k <= K; k <<= 1) {
    for (int j = k >> 1; j > 0; j >>= 1) {
      for (int i = t; i < K; i += nt) {
        int ixj = i ^ j;
        if (ixj > i) {
          unsigned long long a = s[i], b = s[ixj];
          bool up = ((i & k) == 0);
          if (up ? (a < b) : (a > b)) { s[i] = b; s[ixj] = a; }
        }
      }
      __syncthreads();
    }
  }
  for (int i = t; i < K; i += nt) data[i] = s[i];
}

// Decode oriented boxes at selected indices; write clipped/rounded corners +
// score to out_rows, and the AABB (x1,y1,x2,y2,area) for NMS to box.
__global__ void boxes_kernel(const unsigned long long* __restrict__ sel, int K,
                             const float* __restrict__ tblr,
                             const float* __restrict__ orient,  // null => 0
                             const int* __restrict__ swp,
                             const int* __restrict__ shp,
                             const int* __restrict__ Wp,
                             const int* __restrict__ Hp,
                             float* __restrict__ out_rows,
                             float* __restrict__ box) {
  int k = blockIdx.x * blockDim.x + threadIdx.x;
  if (k >= K) return;
  unsigned long long key = sel[k];
  unsigned idx = ~(unsigned)(key & 0xFFFFFFFFull);
  float score = __uint_as_float((unsigned)(key >> 32));
  int yi = (int)(idx >> 9);
  int xi = (int)(idx & 511u);
  float sw = (float)(*swp) * 4.0f;  // stride = 4 for both maps
  float sh = (float)(*shp) * 4.0f;
  float tt = tblr[0 * NPIX + idx], bb = tblr[1 * NPIX + idx];
  float ll = tblr[2 * NPIX + idx], rr = tblr[3 * NPIX + idx];
  float x1 = sw * ((float)xi - ll), x2 = sw * ((float)xi + rr);
  float y1 = sh * ((float)yi - tt), y2 = sh * ((float)yi + bb);
  float ax = sw * (float)xi, ay = sh * (float)yi;
  float th = orient ? orient[idx] : 0.0f;
  float c = cosf(th), s = sinf(th);
  float cx[4] = {x1, x2, x2, x1};
  float cy[4] = {y1, y1, y2, y2};
  float px[4], py[4];
  float xmn = 1e30f, xmx = -1e30f, ymn = 1e30f, ymx = -1e30f;
#pragma unroll
  for (int q = 0; q < 4; ++q) {
    float dx = cx[q] - ax, dy = cy[q] - ay;
    px[q] = ax + dx * c - dy * s;
    py[q] = ay + dx * s + dy * c;
    xmn = fminf(xmn, px[q]); xmx = fmaxf(xmx, px[q]);
    ymn = fminf(ymn, py[q]); ymx = fmaxf(ymx, py[q]);
  }
  box[k * 5 + 0] = xmn; box[k * 5 + 1] = ymn;
  box[k * 5 + 2] = xmx; box[k * 5 + 3] = ymx;
  box[k * 5 + 4] = fmaxf(xmx - xmn, 0.0f) * fmaxf(ymx - ymn, 0.0f);
  float Wm1 = (float)(*Wp) - 1.0f, Hm1 = (float)(*Hp) - 1.0f;
#pragma unroll
  for (int q = 0; q < 4; ++q) {
    out_rows[k * 9 + 2 * q]     = fminf(fmaxf(rintf(px[q]), 0.0f), Wm1);
    out_rows[k * 9 + 2 * q + 1] = fminf(fmaxf(rintf(py[q]), 0.0f), Hm1);
  }
  out_rows[k * 9 + 8] = score;
}

// Suppression-bitmask matrix, one 16x16 (i,j) tile per wave. The pairwise
// denominator term area_i + area_j is produced by one V_WMMA_F32_16X16X4_F32
// (A = [area_i, 1], B = [1 ; area_j] -> C[i][j] = area_i + area_j, exact f32).
// grid is sized exactly so every wave owns a tile (EXEC all ones at the WMMA).
__global__ void supp_kernel(const float* __restrict__ box, int K, int M,
                            float iou_th, unsigned* __restrict__ supp) {
  const int lane = threadIdx.x & 31;
  const int wave = threadIdx.x >> 5;
  const int tile = blockIdx.x * (blockDim.x >> 5) + wave;
  const int tilesPerRow = K >> 4;
  const int ti = tile / tilesPerRow;
  const int tj = tile - ti * tilesPerRow;
  const int ibase = ti << 4, jbase = tj << 4;
  const int li = lane & 15;

  float areai = box[(ibase + li) * 5 + 4];
  float areaj = box[(jbase + li) * 5 + 4];
  float sums[8];
#if HAVE_WMMA_F32X4
  // 32-bit A 16x4 layout: lanes 0-15 hold K=0,1; lanes 16-31 hold K=2,3 (zeros).
  v2f A; A.x = (lane < 16) ? areai : 0.0f; A.y = (lane < 16) ? 1.0f : 0.0f;
  // B 4x16: VGPR0 lanes 0-15 = row K=0 (ones), lanes 16-31 = row K=1 (area_j).
  v2f B; B.x = (lane < 16) ? 1.0f : areaj; B.y = 0.0f;
  v8f C = {};
  C = __builtin_amdgcn_wmma_f32_16x16x4_f32(false, A, false, B, (short)0, C,
                                            false, false);
#pragma unroll
  for (int r = 0; r < 8; ++r) sums[r] = C[r];
#else
  const int mo = (lane < 16) ? 0 : 8;
#pragma unroll
  for (int r = 0; r < 8; ++r)
    sums[r] = box[(ibase + r + mo) * 5 + 4] + areaj;
#endif

  // This lane's column j = jbase + (lane & 15).
  const int j = jbase + li;
  float x1j = box[j * 5 + 0], y1j = box[j * 5 + 1];
  float x2j = box[j * 5 + 2], y2j = box[j * 5 + 3];
  const int mofs = (lane < 16) ? 0 : 8;
  const int w = jbase >> 5;
  const unsigned shl = (unsigned)(jbase & 31);  // 0 or 16
#pragma unroll
  for (int r = 0; r < 8; ++r) {
    int i = ibase + r + mofs;
    bool pred = false;
    if (j > i) {
      float x1i = box[i * 5 + 0], y1i = box[i * 5 + 1];
      float x2i = box[i * 5 + 2], y2i = box[i * 5 + 3];
      float xx1 = fmaxf(x1i, x1j), yy1 = fmaxf(y1i, y1j);
      float xx2 = fminf(x2i, x2j), yy2 = fminf(y2i, y2j);
      float inter = fmaxf(xx2 - xx1, 0.0f) * fmaxf(yy2 - yy1, 0.0f);
      float iou = inter / (sums[r] - inter + 1e-6f);
      pred = iou > iou_th;
    }
    unsigned b32 = (unsigned)__ballot(pred);
    if (lane == 0 && b32) {
      unsigned lo = b32 & 0xFFFFu, hi = b32 >> 16;
      if (lo) atomicOr(&supp[(size_t)(ibase + r) * M + w], lo << shl);
      if (hi) atomicOr(&supp[(size_t)(ibase + r + 8) * M + w], hi << shl);
    }
  }
}

// Greedy sequential NMS scan: one wave, lanes own bitmask words. Reproduces
// the fori_loop semantics exactly; final keep &= (score > 0).
// Rows are prefetched 8 iterations ahead (the scan is latency-bound).
__global__ void nms_scan_kernel(const unsigned* __restrict__ supp,
                                const unsigned long long* __restrict__ sel,
                                int K, float* __restrict__ keep_out) {
  const int lane = threadIdx.x;  // 32 threads
  const int M = K >> 5;
  unsigned rem0 = 0u, rem1 = 0u;  // lane owns words {lane, lane+32}
  for (int i = 0; i < K; ++i) {
    if (i + 8 < K && lane < M)
      __builtin_prefetch(&supp[(size_t)(i + 8) * M + lane], 0, 0);
    const int wrd = i >> 5;
    const int owner = wrd & 31;
    const int slot = wrd >> 5;
    int keptLocal = 0;
    if (lane == owner) {
      unsigned r = slot ? rem1 : rem0;
      keptLocal = ((r >> (i & 31)) & 1u) ? 0 : 1;
    }
    int kept = __shfl(keptLocal, owner, 32);
    if (kept) {
      if (lane < M) rem0 |= supp[(size_t)i * M + lane];
      if (lane + 32 < M) rem1 |= supp[(size_t)i * M + lane + 32];
    }
    if (lane == owner) {
      float sc = __uint_as_float((unsigned)(sel[i] >> 32));
      keep_out[i] = (kept && sc > 0.0f) ? 1.0f : 0.0f;
    }
  }
}

// char_scores[k][c] = cls[c * NPIX + idx_k]
__global__ void gather_cls_kernel(const unsigned long long* __restrict__ csel,
                                  const float* __restrict__ cls,
                                  float* __restrict__ out) {
  int e = blockIdx.x * blockDim.x + threadIdx.x;
  if (e >= KC * NCLS) return;
  int k = e / NCLS, c = e - k * NCLS;
  unsigned idx = ~(unsigned)(csel[k] & 0xFFFFFFFFull);
  const float* p = cls + (size_t)c * NPIX + idx;
  __builtin_prefetch(p, 0, 0);  // global_prefetch_b8
  out[e] = *p;
}

// ---------------- launch ----------------

extern "C" void kernel_launch(void* const* d_in, const int* in_sizes, int n_in,
                              void* d_out, int out_size, void* d_ws,
                              size_t ws_size, hipStream_t stream) {
  const float* wfg    = (const float*)d_in[0];
  const float* wtblr  = (const float*)d_in[1];
  const float* worien = (const float*)d_in[2];
  const float* cfg    = (const float*)d_in[3];
  const float* ctblr  = (const float*)d_in[4];
  const float* ccls   = (const float*)d_in[5];
  const int* swp = (const int*)d_in[6];
  const int* shp = (const int*)d_in[7];
  const int* Wp  = (const int*)d_in[8];
  const int* Hp  = (const int*)d_in[9];
  float* out = (float*)d_out;
  char* ws = (char*)d_ws;

  // workspace layout (bytes)
  unsigned long long* wkeys = (unsigned long long*)(ws + 0);          // 2 MB
  unsigned long long* ckeys = (unsigned long long*)(ws + 2097152);    // 2 MB
  unsigned long long* wsel  = (unsigned long long*)(ws + 4194304);    // 4 KB
  unsigned long long* csel  = (unsigned long long*)(ws + 4198400);    // 16 KB
  unsigned long long* wth   = (unsigned long long*)(ws + 4214784);
  unsigned long long* cth   = (unsigned long long*)(ws + 4214792);
  unsigned* cnt             = (unsigned*)(ws + 4214800);              // 2 u32
  float* wbox               = (float*)(ws + 4214816);                 // 512*5
  float* cbox               = (float*)(ws + 4225056);                 // 2048*5
  unsigned* wsupp           = (unsigned*)(ws + 4266016);              // 512*16
  unsigned* csupp           = (unsigned*)(ws + 4298784);              // 2048*64
  const int n_supp = KW * (KW / 32) + KC * (KC / 32);                 // 139264

  init_kernel<<<(n_supp + 255) / 256, 256, 0, stream>>>(wsupp, n_supp, cnt);
  build_keys_kernel<<<NPIX / 256, 256, 0, stream>>>(wfg, cfg, wkeys, ckeys);

  radix_select_kernel<<<1, 1024, 0, stream>>>(wkeys, NPIX, KW, wth);
  radix_select_kernel<<<1, 1024, 0, stream>>>(ckeys, NPIX, KC, cth);

  compact_kernel<<<NPIX / 256, 256, 0, stream>>>(wkeys, NPIX, wth, wsel, &cnt[0]);
  compact_kernel<<<NPIX / 256, 256, 0, stream>>>(ckeys, NPIX, cth, csel, &cnt[1]);

  bitonic_sort_kernel<<<1, KW / 2, 0, stream>>>(wsel, KW);
  bitonic_sort_kernel<<<1, KC / 2, 0, stream>>>(csel, KC);

  boxes_kernel<<<(KW + 255) / 256, 256, 0, stream>>>(
      wsel, KW, wtblr, worien, swp, shp, Wp, Hp, out + OUT_WORD_BB, wbox);
  boxes_kernel<<<(KC + 255) / 256, 256, 0, stream>>>(
      csel, KC, ctblr, nullptr, swp, shp, Wp, Hp, out + OUT_CHAR_BB, cbox);

  // tiles = (K/16)^2; 8 waves (256 threads) per block; grids divide exactly.
  supp_kernel<<<(KW / 16) * (KW / 16) / 8, 256, 0, stream>>>(
      wbox, KW, KW / 32, 0.3f, wsupp);
  supp_kernel<<<(KC / 16) * (KC / 16) / 8, 256, 0, stream>>>(
      cbox, KC, KC / 32, 0.3f, csupp);

  nms_scan_kernel<<<1, 32, 0, stream>>>(wsupp, wsel, KW, out + OUT_WKEEP);
  nms_scan_kernel<<<1, 32, 0, stream>>>(csupp, csel, KC, out + OUT_CKEEP);

  gather_cls_kernel<<<(KC * NCLS + 255) / 256, 256, 0, stream>>>(
      csel, ccls, out + OUT_CHAR_SC);
}